// MinLSTM_74268574483143
// MI455X (gfx1250) — compile-verified
//
#include <hip/hip_runtime.h>
#include <hip/hip_bf16.h>

typedef __attribute__((ext_vector_type(2))) float v2f;
typedef __attribute__((ext_vector_type(8))) float v8f;

#define B_    8
#define T_    4096
#define IN_   256
#define H_    512
#define OUT_  256
#define ROWS_ (B_ * T_)     // 32768
#define G3_   (3 * H_)      // 1536

// GEMM tiling: block tile 64x128, 8 waves (4 M x 2 N), wave tile 16x64, K chunk 32
#define TM   64
#define TN   128
#define TK   32
#define XSTR 36    // X LDS row stride (floats): 144B -> 16B aligned, bank-step conflict-free
#define WSTR 136   // W LDS row stride (floats): 544B -> 16B aligned, rows k/k+2 on disjoint banks

// scan chunking
#define TCH  128                 // timesteps per chunk
#define NCH  32                  // chunks (T_/TCH)
#define SEQ  (B_ * H_)           // 4096 independent recurrences
#define SEQ2 (SEQ / 2)           // float2-paired sequences

__device__ __forceinline__ float sigmoidf_(float x) { return 1.0f / (1.0f + __expf(-x)); }

// ---------------------------------------------------------------------------
// Kernel 1: gates GEMM fused with activation epilogue, register-buffered
// global->LDS pipeline. Computes f/i/h~ slices with V_WMMA_F32_16X16X4_F32 and
// writes A = sigmoid(f), Bv = sigmoid(i)*h~ (raw 1536-wide gates never hit HBM).
// ---------------------------------------------------------------------------
__global__ __launch_bounds__(256) void gemm_gates_kernel(
    const float* __restrict__ X,    // [ROWS_][IN_]
    const float* __restrict__ W,    // [IN_][G3_]
    float* __restrict__ Ao,         // [ROWS_][H_]
    float* __restrict__ Bvo)        // [ROWS_][H_]
{
    __shared__ float Xs[TM * XSTR];
    __shared__ float Ws0[TK * WSTR];
    __shared__ float Ws1[TK * WSTR];
    __shared__ float Ws2[TK * WSTR];

    const int tid   = threadIdx.x;
    const int lane  = tid & 31;
    const int wave  = tid >> 5;
    const int mb    = blockIdx.x >> 2;        // 512 M blocks
    const int nb    = blockIdx.x & 3;         // 4 N blocks (512/128)
    const int m0    = mb * TM;
    const int n0    = nb * TN;                // in h-space [0, H_)
    const int mw    = (wave >> 1) * 16;       // wave row offset: 0/16/32/48
    const int nw    = (wave & 1) * 64;        // wave col offset: 0/64
    const int l15   = lane & 15;
    const int khalf = (lane >> 4) << 1;       // 0 for lanes 0-15, 2 for lanes 16-31

    // constant staging slots for this thread
    int xr[2], xc[2], wr[4], wc[4];
#pragma unroll
    for (int i = 0; i < 2; i++) { int s = tid + i * 256; xr[i] = s >> 3; xc[i] = (s & 7) * 4; }
#pragma unroll
    for (int i = 0; i < 4; i++) { int s = tid + i * 256; wr[i] = s >> 5; wc[i] = (s & 31) * 4; }

    v8f zero = {};
    v8f accF[4], accI[4], accH[4];
#pragma unroll
    for (int n = 0; n < 4; n++) { accF[n] = zero; accI[n] = zero; accH[n] = zero; }

    float4 rX[2], rW0[4], rW1[4], rW2[4];
    auto load_tiles = [&](int kc) {
#pragma unroll
        for (int i = 0; i < 2; i++)
            rX[i] = *(const float4*)(X + (size_t)(m0 + xr[i]) * IN_ + (kc + xc[i]));
#pragma unroll
        for (int i = 0; i < 4; i++) {
            const float* wb = W + (size_t)(kc + wr[i]) * G3_ + n0 + wc[i];
            rW0[i] = *(const float4*)(wb);
            rW1[i] = *(const float4*)(wb + H_);
            rW2[i] = *(const float4*)(wb + 2 * H_);
        }
    };

    load_tiles(0);
    for (int kc = 0; kc < IN_; kc += TK) {
        // drain staged registers into LDS
#pragma unroll
        for (int i = 0; i < 2; i++) *(float4*)&Xs[xr[i] * XSTR + xc[i]] = rX[i];
#pragma unroll
        for (int i = 0; i < 4; i++) {
            *(float4*)&Ws0[wr[i] * WSTR + wc[i]] = rW0[i];
            *(float4*)&Ws1[wr[i] * WSTR + wc[i]] = rW1[i];
            *(float4*)&Ws2[wr[i] * WSTR + wc[i]] = rW2[i];
        }
        __syncthreads();
        // issue next chunk's global loads; latency hidden under 96 WMMAs below
        if (kc + TK < IN_) load_tiles(kc + TK);

#pragma unroll
        for (int kk = 0; kk < TK / 4; kk++) {
            // A fragment: lanes 0-15 carry K=kk*4+{0,1}, lanes 16-31 K=kk*4+{2,3}
            v2f a = *(const v2f*)&Xs[(mw + l15) * XSTR + kk * 4 + khalf];
            const int kr = kk * 4 + khalf;
#pragma unroll
            for (int n = 0; n < 4; n++) {
                const int col = nw + n * 16 + l15;
                const float* w0 = &Ws0[kr * WSTR + col];
                const float* w1 = &Ws1[kr * WSTR + col];
                const float* w2 = &Ws2[kr * WSTR + col];
                v2f bF, bI, bH;
                bF.x = w0[0]; bF.y = w0[WSTR];
                bI.x = w1[0]; bI.y = w1[WSTR];
                bH.x = w2[0]; bH.y = w2[WSTR];
                accF[n] = __builtin_amdgcn_wmma_f32_16x16x4_f32(
                    false, a, false, bF, (short)0, accF[n], false, false);
                accI[n] = __builtin_amdgcn_wmma_f32_16x16x4_f32(
                    false, a, false, bI, (short)0, accI[n], false, false);
                accH[n] = __builtin_amdgcn_wmma_f32_16x16x4_f32(
                    false, a, false, bH, (short)0, accH[n], false, false);
            }
        }
        __syncthreads();
    }

    // epilogue: A = sigmoid(f), Bv = sigmoid(i) * h~
#pragma unroll
    for (int n = 0; n < 4; n++) {
        const int col = n0 + nw + n * 16 + l15;
#pragma unroll
        for (int r = 0; r < 8; r++) {
            const int row = m0 + mw + ((lane < 16) ? r : r + 8);
            float f  = sigmoidf_(accF[n][r]);
            float it = sigmoidf_(accI[n][r]);
            Ao [(size_t)row * H_ + col] = f;
            Bvo[(size_t)row * H_ + col] = it * accH[n][r];
        }
    }
}

// ---------------------------------------------------------------------------
// Kernel 2a: per-chunk scan summaries. One thread per (sequence-pair, chunk),
// float2 across the h dimension. Reset-aware fold over 128 steps.
// ---------------------------------------------------------------------------
__global__ __launch_bounds__(256) void scan_chunks_kernel(
    const float* __restrict__ A, const float* __restrict__ Bv,
    const int* __restrict__ init,
    float* __restrict__ cP, float* __restrict__ cQ, float* __restrict__ cB)
{
    int g  = blockIdx.x * 256 + threadIdx.x;  // 65536
    int c  = g >> 11;                         // chunk
    int sp = g & (SEQ2 - 1);                  // sequence pair
    int b  = sp >> 8;
    int hp = sp & 255;
    size_t base = ((size_t)b * T_ + (size_t)c * TCH) * H_ + hp * 2;
    const int* ip = init + b * T_ + c * TCH;

    float2 P = make_float2(1.f, 1.f), Q = make_float2(0.f, 0.f);
    float bf = 0.0f;
    for (int t = 0; t < TCH; t++) {
        float2 a = *(const float2*)&A [base + (size_t)t * H_];
        float2 v = *(const float2*)&Bv[base + (size_t)t * H_];
        int r = ip[t];                        // wave-uniform
        if (r) { P = a; Q = v; bf = 1.0f; }
        else {
            Q.x = a.x * Q.x + v.x;  Q.y = a.y * Q.y + v.y;
            P.x = a.x * P.x;        P.y = a.y * P.y;
        }
    }
    size_t o = (size_t)c * SEQ + (size_t)b * H_ + hp * 2;
    *(float2*)&cP[o] = P;
    *(float2*)&cQ[o] = Q;
    *(float2*)&cB[o] = make_float2(bf, bf);
}

// ---------------------------------------------------------------------------
// Kernel 2b: exclusive cross-chunk prefix -> carry-in hidden state per chunk.
// ---------------------------------------------------------------------------
__global__ __launch_bounds__(256) void scan_prefix_kernel(
    const float* __restrict__ cP, const float* __restrict__ cQ,
    const float* __restrict__ cB, const float* __restrict__ h0,
    float* __restrict__ hin)
{
    int sp = blockIdx.x * 256 + threadIdx.x;  // 2048
    size_t so = (size_t)sp * 2;
    float2 h0v = *(const float2*)&h0[so];
    float2 P = make_float2(1.f, 1.f), Q = make_float2(0.f, 0.f);
    for (int c = 0; c < NCH; c++) {
        size_t o = (size_t)c * SEQ + so;
        *(float2*)&hin[o] = make_float2(P.x * h0v.x + Q.x, P.y * h0v.y + Q.y);
        float2 Pc = *(const float2*)&cP[o];
        float2 Qc = *(const float2*)&cQ[o];
        float  bc = cB[o];
        float2 Pn, Qn;
        Pn.x = (bc > 0.f) ? Pc.x : Pc.x * P.x;
        Pn.y = (bc > 0.f) ? Pc.y : Pc.y * P.y;
        Qn.x = (bc > 0.f) ? Qc.x : Pc.x * Q.x + Qc.x;
        Qn.y = (bc > 0.f) ? Qc.y : Pc.y * Q.y + Qc.y;
        P = Pn; Q = Qn;
    }
}

// ---------------------------------------------------------------------------
// Kernel 2c: apply within each chunk: h_t = A_t * (reset ? h0 : h_{t-1}) + B_t
// ---------------------------------------------------------------------------
__global__ __launch_bounds__(256) void scan_apply_kernel(
    const float* __restrict__ A, const float* __restrict__ Bv,
    const int* __restrict__ init, const float* __restrict__ h0,
    const float* __restrict__ hin, float* __restrict__ Hout)
{
    int g  = blockIdx.x * 256 + threadIdx.x;
    int c  = g >> 11;
    int sp = g & (SEQ2 - 1);
    int b  = sp >> 8;
    int hp = sp & 255;
    size_t base = ((size_t)b * T_ + (size_t)c * TCH) * H_ + hp * 2;
    const int* ip = init + b * T_ + c * TCH;

    float2 h0v   = *(const float2*)&h0[(size_t)b * H_ + hp * 2];
    float2 hprev = *(const float2*)&hin[(size_t)c * SEQ + (size_t)b * H_ + hp * 2];
    for (int t = 0; t < TCH; t++) {
        float2 a = *(const float2*)&A [base + (size_t)t * H_];
        float2 v = *(const float2*)&Bv[base + (size_t)t * H_];
        int r = ip[t];
        float px = r ? h0v.x : hprev.x;
        float py = r ? h0v.y : hprev.y;
        hprev.x = a.x * px + v.x;
        hprev.y = a.y * py + v.y;
        *(float2*)&Hout[base + (size_t)t * H_] = hprev;
    }
}

// ---------------------------------------------------------------------------
// Kernel 3: out = h @ W_out  (32768x512)·(512x256), same pipelined WMMA skeleton
// ---------------------------------------------------------------------------
__global__ __launch_bounds__(256) void gemm_out_kernel(
    const float* __restrict__ Hs,   // [ROWS_][H_]
    const float* __restrict__ W,    // [H_][OUT_]
    float* __restrict__ out)        // [ROWS_][OUT_]
{
    __shared__ float Xs[TM * XSTR];
    __shared__ float Ws0[TK * WSTR];

    const int tid   = threadIdx.x;
    const int lane  = tid & 31;
    const int wave  = tid >> 5;
    const int mb    = blockIdx.x >> 1;        // 512 M blocks
    const int nb    = blockIdx.x & 1;         // 2 N blocks (256/128)
    const int m0    = mb * TM;
    const int n0    = nb * TN;
    const int mw    = (wave >> 1) * 16;
    const int nw    = (wave & 1) * 64;
    const int l15   = lane & 15;
    const int khalf = (lane >> 4) << 1;

    int xr[2], xc[2], wr[4], wc[4];
#pragma unroll
    for (int i = 0; i < 2; i++) { int s = tid + i * 256; xr[i] = s >> 3; xc[i] = (s & 7) * 4; }
#pragma unroll
    for (int i = 0; i < 4; i++) { int s = tid + i * 256; wr[i] = s >> 5; wc[i] = (s & 31) * 4; }

    v8f zero = {};
    v8f acc[4];
#pragma unroll
    for (int n = 0; n < 4; n++) acc[n] = zero;

    float4 rX[2], rW[4];
    auto load_tiles = [&](int kc) {
#pragma unroll
        for (int i = 0; i < 2; i++)
            rX[i] = *(const float4*)(Hs + (size_t)(m0 + xr[i]) * H_ + (kc + xc[i]));
#pragma unroll
        for (int i = 0; i < 4; i++)
            rW[i] = *(const float4*)(W + (size_t)(kc + wr[i]) * OUT_ + n0 + wc[i]);
    };

    load_tiles(0);
    for (int kc = 0; kc < H_; kc += TK) {
#pragma unroll
        for (int i = 0; i < 2; i++) *(float4*)&Xs[xr[i] * XSTR + xc[i]] = rX[i];
#pragma unroll
        for (int i = 0; i < 4; i++) *(float4*)&Ws0[wr[i] * WSTR + wc[i]] = rW[i];
        __syncthreads();
        if (kc + TK < H_) load_tiles(kc + TK);

#pragma unroll
        for (int kk = 0; kk < TK / 4; kk++) {
            v2f a = *(const v2f*)&Xs[(mw + l15) * XSTR + kk * 4 + khalf];
            const int kr = kk * 4 + khalf;
#pragma unroll
            for (int n = 0; n < 4; n++) {
                const int col = nw + n * 16 + l15;
                const float* w0 = &Ws0[kr * WSTR + col];
                v2f bb;
                bb.x = w0[0]; bb.y = w0[WSTR];
                acc[n] = __builtin_amdgcn_wmma_f32_16x16x4_f32(
                    false, a, false, bb, (short)0, acc[n], false, false);
            }
        }
        __syncthreads();
    }

#pragma unroll
    for (int n = 0; n < 4; n++) {
        const int col = n0 + nw + n * 16 + l15;
#pragma unroll
        for (int r = 0; r < 8; r++) {
            const int row = m0 + mw + ((lane < 16) ? r : r + 8);
            out[(size_t)row * OUT_ + col] = acc[n][r];
        }
    }
}

// ---------------------------------------------------------------------------
// Kernel 4: h_n = h[:, -1, :]
// ---------------------------------------------------------------------------
__global__ __launch_bounds__(256) void copy_hn_kernel(
    const float* __restrict__ Hs, float* __restrict__ hn)
{
    int i  = blockIdx.x * 256 + threadIdx.x;  // 2048 pairs
    int b  = i >> 8;
    int hp = i & 255;
    *(float2*)&hn[(size_t)b * H_ + hp * 2] =
        *(const float2*)&Hs[((size_t)b * T_ + (T_ - 1)) * H_ + hp * 2];
}

// ---------------------------------------------------------------------------
extern "C" void kernel_launch(void* const* d_in, const int* in_sizes, int n_in,
                              void* d_out, int out_size, void* d_ws, size_t ws_size,
                              hipStream_t stream) {
    const float* X    = (const float*)d_in[0];  // input   (8,4096,256)
    const int*   init = (const int*)  d_in[1];  // is_init (8,4096,1) int32
    const float* h0   = (const float*)d_in[2];  // h_0     (8,1,512)
    const float* Whg  = (const float*)d_in[3];  // W_hg    (256,1536)
    const float* Wout = (const float*)d_in[4];  // W_out   (512,256)

    float* out = (float*)d_out;                       // (8,4096,256)
    float* hn  = out + (size_t)ROWS_ * OUT_;          // (8,1,512)

    float* ws  = (float*)d_ws;
    float* Aa  = ws;                                  // [ROWS_][H_]
    float* Bv  = Aa + (size_t)ROWS_ * H_;             // [ROWS_][H_]
    float* Hs  = Bv + (size_t)ROWS_ * H_;             // [ROWS_][H_]
    float* cP  = Hs + (size_t)ROWS_ * H_;             // [NCH][SEQ]
    float* cQ  = cP + (size_t)NCH * SEQ;
    float* cB  = cQ + (size_t)NCH * SEQ;
    float* hin = cB + (size_t)NCH * SEQ;              // [NCH][SEQ]

    gemm_gates_kernel<<<dim3((ROWS_ / TM) * (H_ / TN)), dim3(256), 0, stream>>>(X, Whg, Aa, Bv);
    scan_chunks_kernel<<<dim3(NCH * SEQ2 / 256), dim3(256), 0, stream>>>(Aa, Bv, init, cP, cQ, cB);
    scan_prefix_kernel<<<dim3(SEQ2 / 256), dim3(256), 0, stream>>>(cP, cQ, cB, h0, hin);
    scan_apply_kernel<<<dim3(NCH * SEQ2 / 256), dim3(256), 0, stream>>>(Aa, Bv, init, h0, hin, Hs);
    gemm_out_kernel<<<dim3((ROWS_ / TM) * (OUT_ / TN)), dim3(256), 0, stream>>>(Hs, Wout, out);
    copy_hn_kernel<<<dim3(SEQ2 / 256), dim3(256), 0, stream>>>(Hs, hn);

    (void)in_sizes; (void)n_in; (void)out_size; (void)ws_size;
}